// FermiNetTorch_15865609191819
// MI455X (gfx1250) — compile-verified
//
#include <hip/hip_runtime.h>
#include <hip/hip_bf16.h>
#include <math.h>

// ---------------------------------------------------------------------------
// FermiNet-like forward pass, MI455X (gfx1250, wave32).
// Memory-bound: ~445 MB of fp32 weights streamed per call -> split-K WMMA
// GEMMs (V_WMMA_F32_16X16X4_F32) with f32 atomics into small accumulators.
// A operands are zero-padded to 16 rows so the inner loop has no masking:
//   2x global_load_b64 + v_wmma per K=4 step.
// ---------------------------------------------------------------------------

typedef float v2f __attribute__((ext_vector_type(2)));
typedef float v8f __attribute__((ext_vector_type(8)));

#define N_E 4
#define N_SV 2048
#define N_PV 1024
#define DL 8192           // 3*N_SV + 2*N_PV
#define D0 56
#define NH 1024           // N_SV/2

// ---------------------------------------------------------------------------
// Split-K GEMM:  acc[m][n] += sum_k A[m][k] * W[n][k]
//   A: 16 x K (row stride lda), rows >= M_store are zero (caller-padded)
//   W: N x K row-major ("x @ w.T" convention)
//   acc: M_store x N, pre-seeded with bias by a prep kernel
// One wave per 16-column tile; grid.y splits K; partials via atomicAdd.
// EXEC stays all-ones (wave-uniform early-out only) as WMMA requires.
// ---------------------------------------------------------------------------
__global__ __launch_bounds__(256)
void gemm_splitk_wmma(const float* __restrict__ A, int lda,
                      const float* __restrict__ W,
                      float* __restrict__ acc,
                      int N, int K, int k_chunk, int M_store)
{
    const int lane = threadIdx.x & 31;
    const int wave = threadIdx.x >> 5;
    const int n0   = (blockIdx.x * 8 + wave) * 16;   // 8 waves per block
    if (n0 >= N) return;                              // wave-uniform

    const int m  = lane & 15;                         // row (A) / col (B) index
    const int kh = (lane >> 4) << 1;                  // 0 for lanes 0-15, 2 for 16-31

    const float* __restrict__ Arow = A + (size_t)m * lda + kh;
    const float* __restrict__ Wrow = W + (size_t)(n0 + m) * K + kh;

    const int k0   = blockIdx.y * k_chunk;
    const int kend = k0 + k_chunk;

    v8f c = {};
    #pragma unroll 8
    for (int k = k0; k < kend; k += 4) {
        // A fragment (16x4 f32): VGPR0 = K{0|2}, VGPR1 = K{1|3} per lane half.
        v2f a = *(const v2f*)(Arow + k);
        // B fragment (4x16 f32): B[k][n] = W[n][k]; same K split per lane half.
        v2f b = *(const v2f*)(Wrow + k);
        c = __builtin_amdgcn_wmma_f32_16x16x4_f32(
                /*neg_a=*/false, a, /*neg_b=*/false, b,
                /*c_mod=*/(short)0, c, /*reuse_a=*/false, /*reuse_b=*/false);
    }

    // C/D layout: VGPR v -> M = v + 8*(lane>=16), N = n0 + (lane&15)
    const int mbase = (lane >> 4) << 3;
    #pragma unroll
    for (int v = 0; v < 8; ++v) {
        int M = mbase + v;
        if (M < M_store)
            atomicAdd(&acc[(size_t)M * N + n0 + m], c[v]);
    }
}

// ---------------------------------------------------------------------------
// Features + layer 0 (tiny: 4x56 @ 56x2048 and 16x4 @ 4x1024). One block.
// ---------------------------------------------------------------------------
__global__ __launch_bounds__(256)
void k_features_layer0(const float* __restrict__ r, const float* __restrict__ a,
                       const float* __restrict__ V0w, const float* __restrict__ V0b,
                       const float* __restrict__ W0w, const float* __restrict__ W0b,
                       float* __restrict__ s_v, float* __restrict__ p_v)
{
    __shared__ float sc[N_E][D0];   // s_cat for layer 0
    __shared__ float pf[16][4];     // pair features (rr, |rr|)

    if (threadIdx.x == 0) {
        float sfeat[N_E][16];
        float pfeat[N_E][N_E][4];
        for (int i = 0; i < N_E; ++i)
            for (int j = 0; j < N_E; ++j) {
                float dx = r[3*i+0] - a[3*j+0];
                float dy = r[3*i+1] - a[3*j+1];
                float dz = r[3*i+2] - a[3*j+2];
                float len = sqrtf(dx*dx + dy*dy + dz*dz);
                sfeat[i][4*j+0] = dx; sfeat[i][4*j+1] = dy;
                sfeat[i][4*j+2] = dz; sfeat[i][4*j+3] = len;
            }
        for (int i = 0; i < N_E; ++i)
            for (int j = 0; j < N_E; ++j) {
                float rx = r[3*j+0] - r[3*i+0];
                float ry = r[3*j+1] - r[3*i+1];
                float rz = r[3*j+2] - r[3*i+2];
                float e  = (i == j) ? 1.0f : 0.0f;
                float lx = rx + e, ly = ry + e, lz = rz + e;
                float len = sqrtf(lx*lx + ly*ly + lz*lz) * (1.0f - e);
                pfeat[i][j][0] = rx; pfeat[i][j][1] = ry;
                pfeat[i][j][2] = rz; pfeat[i][j][3] = len;
            }
        for (int i = 0; i < N_E; ++i) {
            for (int k = 0; k < 16; ++k) {
                sc[i][k]      = 0.5f * (sfeat[0][k] + sfeat[1][k]);
                sc[i][16 + k] = 0.5f * (sfeat[2][k] + sfeat[3][k]);
                sc[i][40 + k] = sfeat[i][k];
            }
            for (int cI = 0; cI < 4; ++cI) {
                sc[i][32 + cI] = 0.5f * (pfeat[0][i][cI] + pfeat[1][i][cI]);
                sc[i][36 + cI] = 0.5f * (pfeat[2][i][cI] + pfeat[3][i][cI]);
            }
        }
        for (int i = 0; i < N_E; ++i)
            for (int j = 0; j < N_E; ++j)
                for (int cI = 0; cI < 4; ++cI)
                    pf[i*4+j][cI] = pfeat[i][j][cI];
    }
    __syncthreads();

    for (int idx = threadIdx.x; idx < N_E * N_SV; idx += blockDim.x) {
        int i = idx >> 11, j = idx & (N_SV - 1);
        float acc = V0b[j];
        const float* wr = V0w + (size_t)j * D0;
        for (int k = 0; k < D0; ++k) acc += sc[i][k] * wr[k];
        s_v[idx] = tanhf(acc);
    }
    for (int idx = threadIdx.x; idx < 16 * N_PV; idx += blockDim.x) {
        int t = idx >> 10, o = idx & (N_PV - 1);
        float acc = W0b[o];
        const float* wr = W0w + (size_t)o * 4;
        for (int cI = 0; cI < 4; ++cI) acc += pf[t][cI] * wr[cI];
        p_v[idx] = tanhf(acc);
    }
}

// ---------------------------------------------------------------------------
// Per-layer prep: build s_cat (16x8192, rows 4..15 zero) and seed the
// accumulators with biases. Grid-strided, multi-block.
// ---------------------------------------------------------------------------
__global__ __launch_bounds__(256)
void k_prep(const float* __restrict__ s_v, const float* __restrict__ p_v,
            const float* __restrict__ Vb, const float* __restrict__ Wb,
            float* __restrict__ s_cat, float* __restrict__ v_acc,
            float* __restrict__ p_acc)
{
    const int tid    = blockIdx.x * blockDim.x + threadIdx.x;
    const int stride = gridDim.x * blockDim.x;

    for (int k = tid; k < N_SV; k += stride) {
        float sfb0 = 0.5f * (s_v[0*N_SV + k] + s_v[1*N_SV + k]);
        float sfb1 = 0.5f * (s_v[2*N_SV + k] + s_v[3*N_SV + k]);
        float vb = Vb[k];
        for (int i = 0; i < N_E; ++i) {
            s_cat[(size_t)i*DL + k]          = sfb0;
            s_cat[(size_t)i*DL + N_SV + k]   = sfb1;
            s_cat[(size_t)i*DL + 3*N_SV + k] = s_v[i*N_SV + k];  // offset 6144
            v_acc[i*N_SV + k] = vb;
        }
    }
    for (int k = tid; k < N_PV; k += stride) {
        float wb = Wb[k];
        for (int j = 0; j < N_E; ++j) {
            float pfb0 = 0.5f * (p_v[(0*4+j)*N_PV + k] + p_v[(1*4+j)*N_PV + k]);
            float pfb1 = 0.5f * (p_v[(2*4+j)*N_PV + k] + p_v[(3*4+j)*N_PV + k]);
            s_cat[(size_t)j*DL + 2*N_SV + k]        = pfb0;  // offset 4096
            s_cat[(size_t)j*DL + 2*N_SV + N_PV + k] = pfb1;  // offset 5120
        }
        for (int t = 0; t < 16; ++t) p_acc[t*N_PV + k] = wb;
    }
    // Zero pad rows 4..15 of s_cat (contiguous 12*DL block).
    for (int idx = tid; idx < 12 * DL; idx += stride)
        s_cat[(size_t)4*DL + idx] = 0.0f;
}

// ---------------------------------------------------------------------------
// Per-layer finalize: s_v = tanh(v_acc); p_v = tanh(p_acc) + p_v (residual).
// ---------------------------------------------------------------------------
__global__ __launch_bounds__(256)
void k_finalize(float* __restrict__ s_v, float* __restrict__ p_v,
                const float* __restrict__ v_acc, const float* __restrict__ p_acc)
{
    int idx = blockIdx.x * blockDim.x + threadIdx.x;
    int stride = gridDim.x * blockDim.x;
    for (int i = idx; i < N_E * N_SV; i += stride)
        s_v[i] = tanhf(v_acc[i]);
    for (int i = idx; i < 16 * N_PV; i += stride)
        p_v[i] = tanhf(p_acc[i]) + p_v[i];
}

// ---------------------------------------------------------------------------
// Head prep: stage zero-padded 16x2048 A operands for the u/d head GEMMs and
// seed the two 2x1024 head accumulators with biases.
// ---------------------------------------------------------------------------
__global__ __launch_bounds__(256)
void k_head_prep(const float* __restrict__ s_v,
                 const float* __restrict__ Vhu_b, const float* __restrict__ Vhd_b,
                 float* __restrict__ A_u, float* __restrict__ A_d,
                 float* __restrict__ acc_u, float* __restrict__ acc_d)
{
    const int tid    = blockIdx.x * blockDim.x + threadIdx.x;
    const int stride = gridDim.x * blockDim.x;

    for (int k = tid; k < 2 * N_SV; k += stride) {
        A_u[k] = s_v[k];               // rows 0-1 = s_v rows 0-1
        A_d[k] = s_v[2*N_SV + k];      // rows 0-1 = s_v rows 2-3
    }
    for (int idx = tid; idx < 14 * N_SV; idx += stride) {
        A_u[2*N_SV + idx] = 0.0f;      // rows 2..15 zero
        A_d[2*N_SV + idx] = 0.0f;
    }
    for (int k = tid; k < NH; k += stride) {
        float bu = Vhu_b[k], bd = Vhd_b[k];
        acc_u[k] = bu; acc_u[NH + k] = bu;
        acc_d[k] = bd; acc_d[NH + k] = bd;
    }
}

// ---------------------------------------------------------------------------
// Head finalize: tanh, 2x2 orbital matrices, envelopes, log|det|+log|det|.
// ---------------------------------------------------------------------------
__global__ __launch_bounds__(256)
void k_head_final(const float* __restrict__ acc_u, const float* __restrict__ acc_d,
                  const float* __restrict__ wu_w, const float* __restrict__ wu_b,
                  const float* __restrict__ wd_w, const float* __restrict__ wd_b,
                  const float* __restrict__ r, const float* __restrict__ a,
                  float* __restrict__ out)
{
    __shared__ float red[8];
    if (threadIdx.x < 8) red[threadIdx.x] = 0.0f;
    __syncthreads();

    for (int h = threadIdx.x; h < NH; h += blockDim.x) {
        for (int i = 0; i < 2; ++i) {
            float su = tanhf(acc_u[i*NH + h]);
            float sd = tanhf(acc_d[i*NH + h]);
            for (int o = 0; o < 2; ++o) {
                atomicAdd(&red[i*2 + o],     su * wu_w[o*NH + h]);
                atomicAdd(&red[4 + i*2 + o], sd * wd_w[o*NH + h]);
            }
        }
    }
    __syncthreads();

    if (threadIdx.x == 0) {
        float fu[2], fd[2];
        for (int i = 0; i < 2; ++i) {
            float su = 0.0f, sd = 0.0f;
            for (int j = 0; j < N_E; ++j) {
                float dx = r[3*i+0] - a[3*j+0];
                float dy = r[3*i+1] - a[3*j+1];
                float dz = r[3*i+2] - a[3*j+2];
                su += expf(-sqrtf(dx*dx + dy*dy + dz*dz));
                float ex = r[3*(i+2)+0] - a[3*j+0];
                float ey = r[3*(i+2)+1] - a[3*j+1];
                float ez = r[3*(i+2)+2] - a[3*j+2];
                sd += expf(-sqrtf(ex*ex + ey*ey + ez*ez));
            }
            fu[i] = su; fd[i] = sd;
        }
        float ou[2][2], od[2][2];
        for (int i = 0; i < 2; ++i)
            for (int o = 0; o < 2; ++o) {
                ou[i][o] = (red[i*2 + o]     + wu_b[o]) * fu[i];
                od[i][o] = (red[4 + i*2 + o] + wd_b[o]) * fd[i];
            }
        float du = ou[0][0]*ou[1][1] - ou[0][1]*ou[1][0];
        float dd = od[0][0]*od[1][1] - od[0][1]*od[1][0];
        out[0] = logf(fabsf(du)) + logf(fabsf(dd));
    }
}

// ---------------------------------------------------------------------------
// Launch sequence (graph-capture safe: only kernel launches on `stream`).
// ---------------------------------------------------------------------------
extern "C" void kernel_launch(void* const* d_in, const int* in_sizes, int n_in,
                              void* d_out, int out_size, void* d_ws, size_t ws_size,
                              hipStream_t stream)
{
    const float* r     = (const float*)d_in[0];
    const float* a     = (const float*)d_in[1];
    const float* V0w   = (const float*)d_in[2];
    const float* V0b   = (const float*)d_in[3];
    const float* Vw    = (const float*)d_in[4];   // (6, 2048, 8192)
    const float* Vb    = (const float*)d_in[5];   // (6, 2048)
    const float* W0w   = (const float*)d_in[6];
    const float* W0b   = (const float*)d_in[7];
    const float* Ww    = (const float*)d_in[8];   // (6, 1024, 1024)
    const float* Wb    = (const float*)d_in[9];   // (6, 1024)
    const float* Vhu_w = (const float*)d_in[10];
    const float* Vhu_b = (const float*)d_in[11];
    const float* Vhd_w = (const float*)d_in[12];
    const float* Vhd_b = (const float*)d_in[13];
    const float* wu_w  = (const float*)d_in[14];
    const float* wu_b  = (const float*)d_in[15];
    const float* wd_w  = (const float*)d_in[16];
    const float* wd_b  = (const float*)d_in[17];

    float* ws     = (float*)d_ws;
    float* s_v    = ws;                 // 4*2048    @ 0
    float* p_v    = ws + 8192;          // 16*1024   @ 8192
    float* s_cat  = ws + 24576;         // 16*8192   @ 24576 (rows 4..15 zero)
    float* v_acc  = ws + 155648;        // 4*2048
    float* p_acc  = ws + 163840;        // 16*1024
    float* A_u    = ws + 180224;        // 16*2048 (rows 2..15 zero)
    float* A_d    = ws + 212992;        // 16*2048 (rows 2..15 zero)
    float* acc_u  = ws + 245760;        // 2*1024
    float* acc_d  = ws + 247808;        // 2*1024

    k_features_layer0<<<1, 256, 0, stream>>>(r, a, V0w, V0b, W0w, W0b, s_v, p_v);

    for (int l = 0; l < 6; ++l) {
        k_prep<<<64, 256, 0, stream>>>(s_v, p_v, Vb + l*N_SV, Wb + l*N_PV,
                                       s_cat, v_acc, p_acc);
        // s_cat(16x8192, 4 live rows) @ Vw[l].T -> v_acc(4x2048)
        // 128 n-tiles x 32 k-chunks = 4096 waves
        gemm_splitk_wmma<<<dim3(16, 32), 256, 0, stream>>>(
            s_cat, DL, Vw + (size_t)l * N_SV * DL, v_acc, N_SV, DL, 256, 4);
        // p_v(16x1024) @ Ww[l].T -> p_acc(16x1024): 64 n-tiles x 8 k-chunks
        gemm_splitk_wmma<<<dim3(8, 8), 256, 0, stream>>>(
            p_v, N_PV, Ww + (size_t)l * N_PV * N_PV, p_acc, N_PV, N_PV, 128, 16);
        k_finalize<<<96, 256, 0, stream>>>(s_v, p_v, v_acc, p_acc);
    }

    k_head_prep<<<32, 256, 0, stream>>>(s_v, Vhu_b, Vhd_b, A_u, A_d, acc_u, acc_d);
    // A_u/A_d(16x2048, 2 live rows) @ Vh{u,d}.T -> acc(2x1024)
    // 64 n-tiles x 16 k-chunks = 1024 waves each
    gemm_splitk_wmma<<<dim3(8, 16), 256, 0, stream>>>(
        A_u, N_SV, Vhu_w, acc_u, NH, N_SV, 128, 2);
    gemm_splitk_wmma<<<dim3(8, 16), 256, 0, stream>>>(
        A_d, N_SV, Vhd_w, acc_d, NH, N_SV, 128, 2);

    k_head_final<<<1, 256, 0, stream>>>(acc_u, acc_d, wu_w, wu_b, wd_w, wd_b,
                                        r, a, (float*)d_out);
}